// VectorQuantizerEMA_87625922773092
// MI455X (gfx1250) — compile-verified
//
#include <hip/hip_runtime.h>

typedef __attribute__((ext_vector_type(2))) float v2f;
typedef __attribute__((ext_vector_type(8))) float v8f;

#define K_CODES 1024
#define DIM     256
#define N_VEC   32768
#define DECAY   0.99f
#define EPSILON 1e-5f

// workspace layout (float offsets)
#define WS_COUNTS 0
#define WS_DW     (WS_COUNTS + K_CODES)
#define WS_E2     (WS_DW + K_CODES * DIM)
#define WS_INVCS  (WS_E2 + K_CODES)

// output layout (float offsets), return order:
// z_q_st, vq_loss, encoding_indices, new_embedding, new_cluster_size, new_ema_w
#define OUT_ZQ      ((size_t)0)
#define OUT_LOSS    (OUT_ZQ + (size_t)N_VEC * DIM)
#define OUT_IDX     (OUT_LOSS + 1)
#define OUT_NEWEMB  (OUT_IDX + N_VEC)
#define OUT_NEWCS   (OUT_NEWEMB + (size_t)K_CODES * DIM)
#define OUT_NEWEMAW (OUT_NEWCS + K_CODES)

// ---------------------------------------------------------------------------
// Zero the scatter accumulators (counts, dw) and the loss scalar.
__global__ void k_zero(float* __restrict__ ws, float* __restrict__ out) {
    int i = blockIdx.x * 256 + threadIdx.x;
    if (i < K_CODES + K_CODES * DIM) ws[WS_COUNTS + i] = 0.0f;   // counts+dw contiguous
    if (i == 0) out[OUT_LOSS] = 0.0f;
}

// ---------------------------------------------------------------------------
// Per-code squared norms ||e_k||^2. One wave per code row (8 waves / block).
__global__ void k_e2(const float* __restrict__ emb, float* __restrict__ e2) {
    int wave = threadIdx.x >> 5;
    int lane = threadIdx.x & 31;
    int row  = blockIdx.x * 8 + wave;
    const float* p = emb + (size_t)row * DIM;
    float s = 0.0f;
    #pragma unroll
    for (int d = lane; d < DIM; d += 32) { float v = p[d]; s += v * v; }
    #pragma unroll
    for (int off = 16; off; off >>= 1) s += __shfl_xor(s, off, 32);
    if (lane == 0) e2[row] = s;
}

// ---------------------------------------------------------------------------
// Main WMMA kernel: argmin_k ||x - e_k||^2 via f32 tensor ops, then gather
// z_q, write z_q_st + indices, accumulate loss, scatter counts/dw.
// Block: 256 threads = 8 waves; each wave owns 16 rows; block covers 128 rows.
__launch_bounds__(256)
__global__ void k_vq(const float* __restrict__ ze, const float* __restrict__ emb,
                     const float* __restrict__ e2, float* __restrict__ out,
                     float* __restrict__ counts, float* __restrict__ dw) {
    // B fragments for one 16-code tile, pre-swizzled: bfrag[t*32 + lane]
    __shared__ v2f bfrag[64 * 32];

    const int tid   = threadIdx.x;
    const int wave  = tid >> 5;
    const int lane  = tid & 31;
    const int rowBase = blockIdx.x * 128 + wave * 16;
    const int mrow  = lane & 15;          // M row within 16 (A) / N col within 16 (B)
    const int khalf = (lane >> 4) * 2;    // K sub-offset per ISA 16x4 f32 A layout

    // A panel resident in registers: a[t] covers K = 4t..4t+3 for this wave's rows
    const float* arow = ze + (size_t)(rowBase + mrow) * DIM + khalf;
    v2f a[64];
    #pragma unroll
    for (int t = 0; t < 64; ++t)
        a[t] = *(const v2f*)(arow + 4 * t);

    float minv[8];
    int   mini[8];
    #pragma unroll
    for (int r = 0; r < 8; ++r) { minv[r] = 3.4e38f; mini[r] = 0; }

    for (int j = 0; j < 64; ++j) {
        const int codeBase = j * 16;
        __syncthreads();
        // Stage embedding tile into LDS in B-fragment order (broadcast to 8 waves).
        for (int i = 0; i < 8; ++i) {
            int e = tid + i * 256;            // 0..2047 fragment-halves
            int t = e >> 5;
            int l = e & 31;
            int n  = l & 15;
            int kh = (l >> 4) * 2;
            const float* src = emb + (size_t)(codeBase + n) * DIM + 4 * t + kh;
            bfrag[e] = *(const v2f*)src;
            if (j < 63)                        // prefetch next tile -> global_prefetch_b8
                __builtin_prefetch(src + DIM * 16, 0, 0);
        }
        __syncthreads();

        // S = X * E^T for this 16x16 tile, K=256 via 64 chained f32 WMMAs.
        v8f acc = {};
        #pragma unroll
        for (int t = 0; t < 64; ++t) {
            acc = __builtin_amdgcn_wmma_f32_16x16x4_f32(
                false, a[t], false, bfrag[t * 32 + lane],
                (short)0, acc, false, false);
        }

        // dist = ||e||^2 - 2 x.e   (||x||^2 constant per row -> dropped for argmin)
        const float e2v  = e2[codeBase + mrow];
        const int   cidx = codeBase + mrow;
        #pragma unroll
        for (int r = 0; r < 8; ++r) {
            float dist = e2v - 2.0f * acc[r];
            if (dist < minv[r]) { minv[r] = dist; mini[r] = cidx; }
        }
    }

    // Cross-lane argmin + gather/scatter epilogue.
    float lsum = 0.0f;
    #pragma unroll
    for (int r = 0; r < 8; ++r) {
        float mv = minv[r];
        int   mi = mini[r];
        // Butterfly within each 16-lane half (masks <= 8 never cross halves).
        #pragma unroll
        for (int off = 8; off; off >>= 1) {
            float ov = __shfl_xor(mv, off, 32);
            int   oi = __shfl_xor(mi, off, 32);
            if (ov < mv || (ov == mv && oi < mi)) { mv = ov; mi = oi; }
        }
        // lanes 0-15 hold result for row rowBase+r, lanes 16-31 for rowBase+r+8
        const int row_lo = rowBase + r;
        const int row_hi = rowBase + r + 8;
        const int idx_lo = __shfl(mi, 0, 32);
        const int idx_hi = __shfl(mi, 16, 32);

        if (lane == 0) {
            out[OUT_IDX + row_lo] = (float)idx_lo;
            out[OUT_IDX + row_hi] = (float)idx_hi;
            atomicAdd(&counts[idx_lo], 1.0f);
            atomicAdd(&counts[idx_hi], 1.0f);
        }

        // z_q gather, z_q_st (== z_q numerically), loss, dw scatter
        const float* zl = ze  + (size_t)row_lo * DIM;
        const float* ql = emb + (size_t)idx_lo * DIM;
        const float* zh = ze  + (size_t)row_hi * DIM;
        const float* qh = emb + (size_t)idx_hi * DIM;
        #pragma unroll
        for (int d = lane; d < DIM; d += 32) {
            float x = zl[d], q = ql[d];
            out[OUT_ZQ + (size_t)row_lo * DIM + d] = q;
            atomicAdd(&dw[(size_t)idx_lo * DIM + d], x);
            float df = q - x; lsum += df * df;

            float x2 = zh[d], q2 = qh[d];
            out[OUT_ZQ + (size_t)row_hi * DIM + d] = q2;
            atomicAdd(&dw[(size_t)idx_hi * DIM + d], x2);
            float df2 = q2 - x2; lsum += df2 * df2;
        }
    }
    // vq_loss = (1 + COMMITMENT_COST) * mean((z_q - z_e)^2)
    #pragma unroll
    for (int off = 16; off; off >>= 1) lsum += __shfl_xor(lsum, off, 32);
    if (lane == 0)
        atomicAdd(&out[OUT_LOSS], lsum * (1.25f / ((float)N_VEC * (float)DIM)));
}

// ---------------------------------------------------------------------------
// EMA cluster sizes + Laplace normalization factor. One block of 1024 threads.
__global__ void k_ema1(const float* __restrict__ ecs, const float* __restrict__ counts,
                       float* __restrict__ out, float* __restrict__ invcs) {
    __shared__ float red[32];
    const int k    = threadIdx.x;
    const int lane = k & 31;
    const int wv   = k >> 5;
    float ncs = DECAY * ecs[k] + (1.0f - DECAY) * counts[k];
    out[OUT_NEWCS + k] = ncs;

    float s = ncs;
    #pragma unroll
    for (int off = 16; off; off >>= 1) s += __shfl_xor(s, off, 32);
    if (lane == 0) red[wv] = s;
    __syncthreads();
    if (k < 32) {
        float t = red[k];
        #pragma unroll
        for (int off = 16; off; off >>= 1) t += __shfl_xor(t, off, 32);
        if (k == 0) red[0] = t;
    }
    __syncthreads();
    const float n = red[0];
    // new_embedding = new_ema_w / ((ncs+eps)/(n+K*eps)*n)
    invcs[k] = (n + (float)K_CODES * EPSILON) / ((ncs + EPSILON) * n);
}

// ---------------------------------------------------------------------------
// new_ema_w and new_embedding.
__global__ void k_ema2(const float* __restrict__ emaw, const float* __restrict__ dw,
                       const float* __restrict__ invcs, float* __restrict__ out) {
    int i = blockIdx.x * 256 + threadIdx.x;
    float w = DECAY * emaw[i] + (1.0f - DECAY) * dw[i];
    out[OUT_NEWEMAW + i] = w;
    out[OUT_NEWEMB + i]  = w * invcs[i >> 8];   // i / DIM
}

// ---------------------------------------------------------------------------
extern "C" void kernel_launch(void* const* d_in, const int* in_sizes, int n_in,
                              void* d_out, int out_size, void* d_ws, size_t ws_size,
                              hipStream_t stream) {
    const float* z_e   = (const float*)d_in[0];   // [32,32,32,256]
    const float* emb   = (const float*)d_in[1];   // [1024,256]
    const float* ecs   = (const float*)d_in[2];   // [1024]
    const float* emaw  = (const float*)d_in[3];   // [1024,256]
    float* out = (float*)d_out;
    float* ws  = (float*)d_ws;

    k_zero<<<(K_CODES + K_CODES * DIM + 255) / 256, 256, 0, stream>>>(ws, out);
    k_e2 <<<K_CODES / 8, 256, 0, stream>>>(emb, ws + WS_E2);
    k_vq <<<N_VEC / 128, 256, 0, stream>>>(z_e, emb, ws + WS_E2, out,
                                           ws + WS_COUNTS, ws + WS_DW);
    k_ema1<<<1, 1024, 0, stream>>>(ecs, ws + WS_COUNTS, out, ws + WS_INVCS);
    k_ema2<<<K_CODES * DIM / 256, 256, 0, stream>>>(emaw, ws + WS_DW,
                                                    ws + WS_INVCS, out);
}